// GCN_5634997092552
// MI455X (gfx1250) — compile-verified
//
#include <hip/hip_runtime.h>
#include <hip/hip_bf16.h>

typedef __attribute__((ext_vector_type(2))) float v2f;
typedef __attribute__((ext_vector_type(8))) float v8f;

__device__ __forceinline__ int imin_i(int a, int b) { return a < b ? a : b; }

// ---------------------------------------------------------------- utilities
__global__ __launch_bounds__(256) void fill_kernel(float* __restrict__ p, float v, int n) {
    int i = blockIdx.x * 256 + threadIdx.x;
    if (i < n) p[i] = v;
}

__global__ __launch_bounds__(256) void deg_accum_kernel(const long long* __restrict__ dst,
                                                        const float* __restrict__ w,
                                                        float* __restrict__ deg, int n_edges) {
    int e = blockIdx.x * 256 + threadIdx.x;
    if (e >= n_edges) return;
    atomicAdd(&deg[(int)__builtin_nontemporal_load(dst + e)],
              __builtin_nontemporal_load(w + e));
}

__global__ __launch_bounds__(256) void rsqrt_kernel(float* __restrict__ p, int n) {
    int i = blockIdx.x * 256 + threadIdx.x;
    if (i < n) p[i] = rsqrtf(p[i]);   // deg >= 1 (self loop), no zero guard needed
}

__global__ __launch_bounds__(256) void norm_kernel(const long long* __restrict__ src,
                                                   const long long* __restrict__ dst,
                                                   const float* __restrict__ w,
                                                   const float* __restrict__ dinv,
                                                   float* __restrict__ nrm, int n_edges) {
    int e = blockIdx.x * 256 + threadIdx.x;
    if (e >= n_edges) return;
    const int s = (int)__builtin_nontemporal_load(src + e);
    const int d = (int)__builtin_nontemporal_load(dst + e);
    nrm[e] = dinv[s] * __builtin_nontemporal_load(w + e) * dinv[d];
}

// ------------------------------------------------- dense transform via WMMA
// h[N,OUT] = x[N,IN] @ W[IN,OUT]   (bias handled later, fused with aggregation)
// One wave handles a 16-node tile. V_WMMA_F32_16X16X4_F32 layouts:
//   A (16x4): lane l -> row m=l%16, K pair base kb=(l/16)*2 -> {A[m][kb], A[m][kb+1]}
//   B (4x16): lane l -> col n=l%16, K pair base kb=(l/16)*2 -> {B[kb][n], B[kb+1][n]}
//   C/D (16x16): lane l col n=l%16; vgpr j -> row j + 8*(l/16)
template <int IN, int OUT>
__global__ __launch_bounds__(256) void transform_wmma(const float* __restrict__ x,
                                                      const float* __restrict__ W,
                                                      float* __restrict__ h, int n_nodes) {
    constexpr bool FULLN = (OUT % 16 == 0);       // N-tile never masked
    const int lane = threadIdx.x & 31;
    const int wid  = threadIdx.x >> 5;
    const int tile = blockIdx.x * 8 + wid;
    const int row0 = tile * 16;
    if (row0 >= n_nodes) return;                  // wave-uniform exit, EXEC stays full

    const int m  = lane & 15;
    const int kb = (lane >> 4) * 2;               // K pair base for this half-wave
    const bool full_tile = (row0 + 16 <= n_nodes);   // wave-uniform
    const int node = full_tile ? (row0 + m) : imin_i(row0 + m, n_nodes - 1);
    const int rbase = row0 + (lane >> 4) * 8;

#pragma unroll
    for (int nt = 0; nt < OUT; nt += 16) {
        v8f c = {};
        const int col = nt + m;
        const bool colok = FULLN ? true : (col < OUT);
#pragma unroll
        for (int kk = 0; kk < IN; kk += 4) {
            const int ka = kk + kb;
            v2f a, b;
            a.x = x[(size_t)node * IN + ka];      // contiguous pair -> global_load_b64
            a.y = x[(size_t)node * IN + ka + 1];
            b.x = colok ? W[(size_t)ka * OUT + col] : 0.0f;
            b.y = colok ? W[(size_t)(ka + 1) * OUT + col] : 0.0f;
            c = __builtin_amdgcn_wmma_f32_16x16x4_f32(false, a, false, b,
                                                      (short)0, c, false, false);
        }
        if (full_tile) {
            // branch-free: half-wave j-stores cover 16 contiguous floats of a row
            if (colok) {
#pragma unroll
                for (int j = 0; j < 8; ++j)
                    h[(size_t)(rbase + j) * OUT + col] = c[j];
            }
        } else {
            if (colok) {
#pragma unroll
                for (int j = 0; j < 8; ++j) {
                    const int r = rbase + j;
                    if (r < n_nodes) h[(size_t)r * OUT + col] = c[j];
                }
            }
        }
    }
}

// ------------------------------------------------------------- edge scatter
// agg[dst] += h[src] * norm  (one thread per edge, float4 gathers, f32 atomics)
template <int OUT>
__global__ __launch_bounds__(256) void scatter_edges(const long long* __restrict__ src,
                                                     const long long* __restrict__ dst,
                                                     const float* __restrict__ nrm,
                                                     const float* __restrict__ h,
                                                     float* __restrict__ agg, int n_edges) {
    int e = blockIdx.x * 256 + threadIdx.x;
    if (e >= n_edges) return;
    __builtin_prefetch(src + e + 2048, 0, 0);     // global_prefetch_b8 ahead-stream
    __builtin_prefetch(dst + e + 2048, 0, 0);
    const int s = (int)__builtin_nontemporal_load(src + e);   // single-use streams: TH_NT
    const int d = (int)__builtin_nontemporal_load(dst + e);
    const float w = __builtin_nontemporal_load(nrm + e);
    const float* __restrict__ hs = h + (size_t)s * OUT;
    float* __restrict__ ad = agg + (size_t)d * OUT;
    if constexpr (OUT >= 4) {
#pragma unroll
        for (int f = 0; f < OUT; f += 4) {
            float4 v = *(const float4*)(hs + f);
            atomicAdd(ad + f + 0, v.x * w);
            atomicAdd(ad + f + 1, v.y * w);
            atomicAdd(ad + f + 2, v.z * w);
            atomicAdd(ad + f + 3, v.w * w);
        }
    } else {
#pragma unroll
        for (int f = 0; f < OUT; ++f) atomicAdd(ad + f, hs[f] * w);
    }
}

// ------------------------------------------- fused self-loop + bias (+ReLU)
// h[n][f] = act( agg[n][f] + h[n][f]*dinv[n]^2 + b[f] )   (in place on h)
template <int OUT, bool RELU>
__global__ __launch_bounds__(256) void finalize_kernel(float* __restrict__ h,
                                                       const float* __restrict__ agg,
                                                       const float* __restrict__ dinv,
                                                       const float* __restrict__ bias,
                                                       int n_nodes) {
    int i = blockIdx.x * 256 + threadIdx.x;
    if (i >= n_nodes * OUT) return;
    const int n = i / OUT;                        // OUT is 2/16/32 -> shifts
    const int f = i - n * OUT;
    const float s = dinv[n];
    float v = agg[i] + h[i] * (s * s) + bias[f];
    if (RELU) v = fmaxf(v, 0.0f);
    h[i] = v;
}

// ------------------------------------------------------------------ pooling
__global__ __launch_bounds__(256) void pool_accum_kernel(const float* __restrict__ h,
                                                         const long long* __restrict__ batch,
                                                         float* __restrict__ sums,
                                                         float* __restrict__ cnt, int n_nodes) {
    int n = blockIdx.x * 256 + threadIdx.x;
    if (n >= n_nodes) return;
    const int g = (int)__builtin_nontemporal_load(batch + n);
    atomicAdd(&sums[g * 2 + 0], h[(size_t)n * 2 + 0]);
    atomicAdd(&sums[g * 2 + 1], h[(size_t)n * 2 + 1]);
    atomicAdd(&cnt[g], 1.0f);
}

__global__ __launch_bounds__(128) void pool_div_kernel(const float* __restrict__ sums,
                                                       const float* __restrict__ cnt,
                                                       float* __restrict__ out, int n_out) {
    int i = threadIdx.x;
    if (i < n_out) out[i] = sums[i] / fmaxf(cnt[i >> 1], 1.0f);
}

// ------------------------------------------------------------------- driver
extern "C" void kernel_launch(void* const* d_in, const int* in_sizes, int n_in,
                              void* d_out, int out_size, void* d_ws, size_t ws_size,
                              hipStream_t stream) {
    const float*     x      = (const float*)d_in[0];       // [N,4]
    const long long* eidx   = (const long long*)d_in[1];   // [2,E]
    const float*     ew     = (const float*)d_in[2];       // [E]
    const long long* batch  = (const long long*)d_in[3];   // [N]
    const float *W1 = (const float*)d_in[4], *b1 = (const float*)d_in[5];
    const float *W2 = (const float*)d_in[6], *b2 = (const float*)d_in[7];
    const float *W3 = (const float*)d_in[8], *b3 = (const float*)d_in[9];
    float* out = (float*)d_out;

    const int n_nodes = in_sizes[0] / 4;
    const int n_edges = in_sizes[2];
    const long long* src = eidx;             // edge_index[0]
    const long long* dst = eidx + n_edges;   // edge_index[1]

    // workspace carve-up (256B aligned)
    size_t off = 0;
    auto carve = [&](size_t bytes) {
        size_t r = off;
        off += (bytes + 255) & ~(size_t)255;
        return r;
    };
    char* ws = (char*)d_ws;
    float* dinv  = (float*)(ws + carve((size_t)n_nodes * 4));        // deg -> dinv
    float* nrm   = (float*)(ws + carve((size_t)n_edges * 4));
    float* bufP  = (float*)(ws + carve((size_t)n_nodes * 32 * 4));
    float* bufQ  = (float*)(ws + carve((size_t)n_nodes * 32 * 4));
    float* agg   = (float*)(ws + carve((size_t)n_nodes * 32 * 4));
    float* psums = (float*)(ws + carve(64 * 2 * 4));
    float* pcnt  = (float*)(ws + carve(64 * 4));

    auto blks = [](long long n) { return (int)((n + 255) / 256); };
    const int ebl = blks(n_edges);
    const int tile_blocks = (int)(((n_nodes + 15) / 16 + 7) / 8);   // 8 waves/block

    // --- normalization: deg (self loop weight 1) -> dinv -> per-edge norm
    fill_kernel<<<blks(n_nodes), 256, 0, stream>>>(dinv, 1.0f, n_nodes);
    deg_accum_kernel<<<ebl, 256, 0, stream>>>(dst, ew, dinv, n_edges);
    rsqrt_kernel<<<blks(n_nodes), 256, 0, stream>>>(dinv, n_nodes);
    norm_kernel<<<ebl, 256, 0, stream>>>(src, dst, ew, dinv, nrm, n_edges);

    // --- layer 1: 4 -> 16, ReLU
    transform_wmma<4, 16><<<tile_blocks, 256, 0, stream>>>(x, W1, bufP, n_nodes);
    fill_kernel<<<blks((long long)n_nodes * 16), 256, 0, stream>>>(agg, 0.0f, n_nodes * 16);
    scatter_edges<16><<<ebl, 256, 0, stream>>>(src, dst, nrm, bufP, agg, n_edges);
    finalize_kernel<16, true><<<blks((long long)n_nodes * 16), 256, 0, stream>>>(
        bufP, agg, dinv, b1, n_nodes);

    // --- layer 2: 16 -> 32, ReLU
    transform_wmma<16, 32><<<tile_blocks, 256, 0, stream>>>(bufP, W2, bufQ, n_nodes);
    fill_kernel<<<blks((long long)n_nodes * 32), 256, 0, stream>>>(agg, 0.0f, n_nodes * 32);
    scatter_edges<32><<<ebl, 256, 0, stream>>>(src, dst, nrm, bufQ, agg, n_edges);
    finalize_kernel<32, true><<<blks((long long)n_nodes * 32), 256, 0, stream>>>(
        bufQ, agg, dinv, b2, n_nodes);

    // --- layer 3: 32 -> 2, no activation
    transform_wmma<32, 2><<<tile_blocks, 256, 0, stream>>>(bufQ, W3, bufP, n_nodes);
    fill_kernel<<<blks((long long)n_nodes * 2), 256, 0, stream>>>(agg, 0.0f, n_nodes * 2);
    scatter_edges<2><<<ebl, 256, 0, stream>>>(src, dst, nrm, bufP, agg, n_edges);
    finalize_kernel<2, false><<<blks((long long)n_nodes * 2), 256, 0, stream>>>(
        bufP, agg, dinv, b3, n_nodes);

    // --- global mean pool
    fill_kernel<<<1, 256, 0, stream>>>(psums, 0.0f, 192);   // sums(128) + cnt(64) contiguous
    pool_accum_kernel<<<blks(n_nodes), 256, 0, stream>>>(bufP, batch, psums, pcnt, n_nodes);
    pool_div_kernel<<<1, 128, 0, stream>>>(psums, pcnt, out, out_size);
}